// MambaClassifier_37958920962160
// MI455X (gfx1250) — compile-verified
//
#include <hip/hip_runtime.h>
#include <math.h>

// ---------------------------------------------------------------------------
// MambaClassifier forward for gfx1250 (CDNA5, wave32).
// All dense layers use V_WMMA_F32_16X16X4_F32 (full fp32 matrix core path).
// Each wave computes a 32x64 output tile (8 accumulators) for B-reuse.
// Sequential selective-scan runs in registers, parallel over (B x d-chunks).
// ---------------------------------------------------------------------------

#define BATCH 32
#define LSEQ  256          // (224/14)^2
#define DM_   384
#define DIN_  768
#define NST_  16
#define RLOW  48
#define KCONV 4
#define NC_   1000
#define KPATCH 588         // 3*14*14
#define ROWS  (BATCH * LSEQ)        // 8192 token rows

typedef float v2f __attribute__((ext_vector_type(2)));
typedef float v8f __attribute__((ext_vector_type(8)));

__device__ __forceinline__ v8f wmma_f32(v2f a, v2f b, v8f c) {
    // D = A(16x4) * B(4x16) + C(16x16), fp32, wave32
    return __builtin_amdgcn_wmma_f32_16x16x4_f32(false, a, false, b, (short)0, c,
                                                 false, false);
}

// ---------------------------------------------------------------------------
// Generic GEMM: D[m,n] = sum_k A[m*lda+k] * W[n*K+k]   (weights row-major (N,K))
// Wave tile: 32 rows x 64 cols. M must be a multiple of 32.
// EPI: 0 = none, 1 = +bias, 2 = +bias then softplus, 3 = +residual
// ---------------------------------------------------------------------------
template <int EPI>
__global__ void gemm_xwT(const float* __restrict__ A, const float* __restrict__ W,
                         float* __restrict__ D, const float* __restrict__ bias,
                         const float* __restrict__ res, int ldres,
                         int M, int N, int K, int lda, int ldd) {
    const int NQ   = (N + 63) >> 6;                       // 64-wide column quads
    const int wid  = (blockIdx.x * blockDim.x + threadIdx.x) >> 5;
    const int lane = threadIdx.x & 31;
    const int mt = wid / NQ, nq = wid - mt * NQ;
    if (mt * 32 >= M) return;                             // whole-wave uniform
    const int m0 = mt * 32, n0 = nq * 64;
    const int khalf = (lane >> 4) * 2;                    // K sub-slot per lane half
    const int ncol  = n0 + (lane & 15);

    const float* Ap0 = A + (size_t)(m0 + (lane & 15)) * lda + khalf;
    const float* Ap1 = Ap0 + (size_t)16 * lda;
    int c0i = ncol,      c1i = ncol + 16, c2i = ncol + 32, c3i = ncol + 48;
    c0i = c0i < N ? c0i : N - 1;  c1i = c1i < N ? c1i : N - 1;   // clamp (cols are
    c2i = c2i < N ? c2i : N - 1;  c3i = c3i < N ? c3i : N - 1;   // independent)
    const float* W0 = W + (size_t)c0i * K + khalf;
    const float* W1 = W + (size_t)c1i * K + khalf;
    const float* W2 = W + (size_t)c2i * K + khalf;
    const float* W3 = W + (size_t)c3i * K + khalf;

    v8f acc[2][4] = {};
    for (int k = 0; k < K; k += 4) {
        v2f a0 = *(const v2f*)(Ap0 + k);
        v2f a1 = *(const v2f*)(Ap1 + k);
        v2f b0 = *(const v2f*)(W0 + k);
        v2f b1 = *(const v2f*)(W1 + k);
        v2f b2 = *(const v2f*)(W2 + k);
        v2f b3 = *(const v2f*)(W3 + k);
        acc[0][0] = wmma_f32(a0, b0, acc[0][0]);
        acc[0][1] = wmma_f32(a0, b1, acc[0][1]);
        acc[0][2] = wmma_f32(a0, b2, acc[0][2]);
        acc[0][3] = wmma_f32(a0, b3, acc[0][3]);
        acc[1][0] = wmma_f32(a1, b0, acc[1][0]);
        acc[1][1] = wmma_f32(a1, b1, acc[1][1]);
        acc[1][2] = wmma_f32(a1, b2, acc[1][2]);
        acc[1][3] = wmma_f32(a1, b3, acc[1][3]);
    }

    for (int mi = 0; mi < 2; ++mi) {
        const int rbase = m0 + mi * 16 + (lane >> 4) * 8; // D rows for this lane half
        for (int q = 0; q < 4; ++q) {
            v8f c = acc[mi][q];
            int n = n0 + q * 16 + (lane & 15);
            if (n >= N) continue;                         // post-WMMA divergence OK
            float bv = (EPI == 1 || EPI == 2) ? bias[n] : 0.f;
            for (int r = 0; r < 8; ++r) {
                int row = rbase + r;
                float v = c[r] + bv;
                if (EPI == 2) v = fmaxf(v, 0.f) + log1pf(__expf(-fabsf(v)));  // softplus
                if (EPI == 3) v += res[(size_t)row * ldres + n];
                D[(size_t)row * ldd + n] = v;
            }
        }
    }
}

// ---------------------------------------------------------------------------
// im2col for the patch embedding: out[m, k] = x[b, c, py*14+i, px*14+j]
// m = b*256 + py*16 + px, k = c*196 + i*14 + j. One-shot, bandwidth-bound.
// ---------------------------------------------------------------------------
__global__ void im2col_patches(const float* __restrict__ x, float* __restrict__ col) {
    int idx = blockIdx.x * blockDim.x + threadIdx.x;
    if (idx >= ROWS * KPATCH) return;
    int m = idx / KPATCH, k = idx - m * KPATCH;
    int b = m >> 8, pidx = m & 255;
    int py = pidx >> 4, px = pidx & 15;
    int c = k / 196, rem = k - c * 196;
    int i = rem / 14, j = rem - i * 14;
    col[idx] = x[(size_t)(b * 3 + c) * 224 * 224 + (size_t)(py * 14 + i) * 224 +
                 (px * 14 + j)];
}

// ---------------------------------------------------------------------------
// LayerNorm over last dim (384). One wave32 per token row, 12 elems/lane.
// ---------------------------------------------------------------------------
__global__ void layernorm_k(const float* __restrict__ in, const float* __restrict__ w,
                            const float* __restrict__ bias, float* __restrict__ out,
                            int rows) {
    const int wid  = (blockIdx.x * blockDim.x + threadIdx.x) >> 5;
    const int lane = threadIdx.x & 31;
    if (wid >= rows) return;
    const float* r = in + (size_t)wid * DM_;
    float v[12];
    float s = 0.f;
    for (int i = 0; i < 12; ++i) { v[i] = r[lane + i * 32]; s += v[i]; }
    for (int o = 16; o > 0; o >>= 1) s += __shfl_xor(s, o, 32);
    float mean = s * (1.f / DM_);
    float sq = 0.f;
    for (int i = 0; i < 12; ++i) { float d = v[i] - mean; sq += d * d; }
    for (int o = 16; o > 0; o >>= 1) sq += __shfl_xor(sq, o, 32);
    float inv = rsqrtf(sq * (1.f / DM_) + 1e-5f);
    float* op = out + (size_t)wid * DM_;
    for (int i = 0; i < 12; ++i) {
        int d = lane + i * 32;
        op[d] = (v[i] - mean) * inv * w[d] + bias[d];
    }
}

// ---------------------------------------------------------------------------
// Causal depthwise conv1d (K=4, left pad 3) + SiLU.  Input = first half of xz.
// ---------------------------------------------------------------------------
__global__ void conv_silu(const float* __restrict__ xz, const float* __restrict__ cw,
                          const float* __restrict__ cb, float* __restrict__ xb) {
    int idx = blockIdx.x * blockDim.x + threadIdx.x;
    if (idx >= ROWS * DIN_) return;
    int d = idx % DIN_;
    int bl = idx / DIN_;
    int l = bl & (LSEQ - 1), b = bl >> 8;
    const float* w = cw + d * KCONV;
    float acc = cb[d];
    for (int j = 0; j < KCONV; ++j) {
        int ls = l + j - (KCONV - 1);
        if (ls >= 0) acc += w[j] * xz[(size_t)(b * LSEQ + ls) * (2 * DIN_) + d];
    }
    xb[(size_t)bl * DIN_ + d] = acc / (1.f + __expf(-acc));   // silu
}

// ---------------------------------------------------------------------------
// Selective SSM scan. Grid: B * (DIN/128) blocks of 256 threads.
// Thread pair (d even/odd lanes) owns one channel: 8 states each in registers.
// Reads x from xb and writes y back in place; dt is strided inside xz buffer.
// ---------------------------------------------------------------------------
__global__ void ssm_scan(const float* __restrict__ A_log, const float* __restrict__ Dp,
                         const float* __restrict__ dtbuf,  // stride 1536
                         const float* __restrict__ proj,   // stride 80: [dt|B|C]
                         float* __restrict__ xb) {         // x in, y out (stride 768)
    const int b  = blockIdx.x / 6;
    const int d0 = (blockIdx.x % 6) * 128;
    const int t  = threadIdx.x;
    const int d  = d0 + (t >> 1);
    const int nb = (t & 1) * 8;
    float Ar[8], h[8];
    for (int j = 0; j < 8; ++j) {
        Ar[j] = -__expf(A_log[(size_t)d * NST_ + nb + j]);
        h[j]  = 0.f;
    }
    const float Dd = Dp[d];
    const size_t rb = (size_t)b * LSEQ;
    for (int l = 0; l < LSEQ; ++l) {
        size_t row = rb + l;
        float dtv = dtbuf[row * (2 * DIN_) + d];
        float xv  = xb[row * DIN_ + d];
        const float* pr = proj + row * (RLOW + 2 * NST_);
        float4 B0 = *(const float4*)(pr + RLOW + nb);
        float4 B1 = *(const float4*)(pr + RLOW + nb + 4);
        float4 C0 = *(const float4*)(pr + RLOW + NST_ + nb);
        float4 C1 = *(const float4*)(pr + RLOW + NST_ + nb + 4);
        float Bb[8] = {B0.x, B0.y, B0.z, B0.w, B1.x, B1.y, B1.z, B1.w};
        float Cc[8] = {C0.x, C0.y, C0.z, C0.w, C1.x, C1.y, C1.z, C1.w};
        float dx = dtv * xv, acc = 0.f;
        for (int j = 0; j < 8; ++j) {
            h[j] = h[j] * __expf(dtv * Ar[j]) + dx * Bb[j];
            acc += h[j] * Cc[j];
        }
        acc += __shfl_xor(acc, 1, 32);                    // reduce the two n-halves
        if ((t & 1) == 0) xb[row * DIN_ + d] = acc + xv * Dd;
    }
}

// y *= silu(z), z = second half of xz row
__global__ void gate_silu(float* __restrict__ y, const float* __restrict__ xz) {
    int idx = blockIdx.x * blockDim.x + threadIdx.x;
    if (idx >= ROWS * DIN_) return;
    int d = idx % DIN_, r = idx / DIN_;
    float z = xz[(size_t)r * (2 * DIN_) + DIN_ + d];
    y[idx] *= z / (1.f + __expf(-z));
}

// mean over L for each (b, dm)
__global__ void pool_mean(const float* __restrict__ in, float* __restrict__ out) {
    int idx = blockIdx.x * blockDim.x + threadIdx.x;
    if (idx >= BATCH * DM_) return;
    int b = idx / DM_, d = idx % DM_;
    float s = 0.f;
    for (int l = 0; l < LSEQ; ++l) s += in[(size_t)(b * LSEQ + l) * DM_ + d];
    out[idx] = s * (1.f / LSEQ);
}

static inline int gemm_blocks(int M, int N) {
    int waves = (M / 32) * ((N + 63) / 64);
    return (waves + 7) / 8;                               // 8 waves / 256-thread block
}

extern "C" void kernel_launch(void* const* d_in, const int* in_sizes, int n_in,
                              void* d_out, int out_size, void* d_ws, size_t ws_size,
                              hipStream_t stream) {
    const float* x        = (const float*)d_in[0];
    const float* patch_w  = (const float*)d_in[1];
    const float* patch_b  = (const float*)d_in[2];
    const float* norm_w   = (const float*)d_in[3];
    const float* norm_b   = (const float*)d_in[4];
    const float* ipw      = (const float*)d_in[5];
    const float* conv_w   = (const float*)d_in[6];
    const float* conv_b   = (const float*)d_in[7];
    const float* A_log    = (const float*)d_in[8];
    const float* D_ssm    = (const float*)d_in[9];
    const float* xproj_w  = (const float*)d_in[10];
    const float* dtproj_w = (const float*)d_in[11];
    const float* dtproj_b = (const float*)d_in[12];
    const float* opw      = (const float*)d_in[13];
    const float* fnorm_w  = (const float*)d_in[14];
    const float* fnorm_b  = (const float*)d_in[15];
    const float* cls_w    = (const float*)d_in[16];
    const float* cls_b    = (const float*)d_in[17];
    float* out = (float*)d_out;

    // workspace layout (floats); total ~123 MB
    float* ws     = (float*)d_ws;
    float* t      = ws;                                   // 8192*384
    float* xn     = t + (size_t)ROWS * DM_;               // 8192*384
    float* xz     = xn + (size_t)ROWS * DM_;              // 8192*1536 (dt reuses cols 0..767)
    float* xb     = xz + (size_t)ROWS * 2 * DIN_;         // 8192*768 (y in place)
    float* proj   = xb + (size_t)ROWS * DIN_;             // 8192*80
    float* pooled = proj + (size_t)ROWS * (RLOW + 2 * NST_);  // 32*384
    float* col    = pooled + (size_t)BATCH * DM_;         // 8192*588 im2col

    const int EW = 256;
    const int nElemDin = ROWS * DIN_;

    // patch embedding: im2col then WMMA GEMM (+bias) -> t (B, 256, 384)
    im2col_patches<<<(ROWS * KPATCH + EW - 1) / EW, EW, 0, stream>>>(x, col);
    gemm_xwT<1><<<gemm_blocks(ROWS, DM_), EW, 0, stream>>>(
        col, patch_w, t, patch_b, nullptr, 0, ROWS, DM_, KPATCH, KPATCH, DM_);

    for (int l = 0; l < 4; ++l) {
        // 1) LayerNorm
        layernorm_k<<<ROWS / 8, EW, 0, stream>>>(t, norm_w + l * DM_, norm_b + l * DM_,
                                                 xn, ROWS);
        // 2) in_proj: (8192x384) @ (1536x384)^T -> xz
        gemm_xwT<0><<<gemm_blocks(ROWS, 2 * DIN_), EW, 0, stream>>>(
            xn, ipw + (size_t)l * 2 * DIN_ * DM_, xz, nullptr, nullptr, 0,
            ROWS, 2 * DIN_, DM_, DM_, 2 * DIN_);
        // 3) causal depthwise conv + silu -> xb
        conv_silu<<<(nElemDin + EW - 1) / EW, EW, 0, stream>>>(
            xz, conv_w + (size_t)l * DIN_ * KCONV, conv_b + l * DIN_, xb);
        // 4) x_proj: (8192x768) @ (80x768)^T -> proj [dt_low | B | C]
        gemm_xwT<0><<<gemm_blocks(ROWS, RLOW + 2 * NST_), EW, 0, stream>>>(
            xb, xproj_w + (size_t)l * (RLOW + 2 * NST_) * DIN_, proj, nullptr, nullptr, 0,
            ROWS, RLOW + 2 * NST_, DIN_, DIN_, RLOW + 2 * NST_);
        // 5) dt_proj + bias + softplus -> dt (stored into xz cols 0..767, stride 1536)
        gemm_xwT<2><<<gemm_blocks(ROWS, DIN_), EW, 0, stream>>>(
            proj, dtproj_w + (size_t)l * DIN_ * RLOW, xz, dtproj_b + l * DIN_, nullptr, 0,
            ROWS, DIN_, RLOW, RLOW + 2 * NST_, 2 * DIN_);
        // 6) selective scan (y overwrites xb)
        ssm_scan<<<BATCH * (DIN_ / 128), EW, 0, stream>>>(
            A_log + (size_t)l * DIN_ * NST_, D_ssm + l * DIN_, xz, proj, xb);
        // 7) gate with silu(z)
        gate_silu<<<(nElemDin + EW - 1) / EW, EW, 0, stream>>>(xb, xz);
        // 8) out_proj + residual -> t
        gemm_xwT<3><<<gemm_blocks(ROWS, DM_), EW, 0, stream>>>(
            xb, opw + (size_t)l * DM_ * DIN_, t, nullptr, t, DM_,
            ROWS, DM_, DIN_, DIN_, DM_);
    }

    // final LN -> xn, mean pool -> pooled, classifier -> out
    layernorm_k<<<ROWS / 8, EW, 0, stream>>>(t, fnorm_w, fnorm_b, xn, ROWS);
    pool_mean<<<(BATCH * DM_ + EW - 1) / EW, EW, 0, stream>>>(xn, pooled);
    gemm_xwT<1><<<gemm_blocks(BATCH, NC_), EW, 0, stream>>>(
        pooled, cls_w, out, cls_b, nullptr, 0, BATCH, NC_, DM_, DM_, NC_);
}